// GCN_29231547416806
// MI455X (gfx1250) — compile-verified
//
#include <hip/hip_runtime.h>

typedef __attribute__((ext_vector_type(2))) float v2f;
typedef __attribute__((ext_vector_type(8))) float v8f;
typedef __attribute__((ext_vector_type(4))) int   v4i;

__device__ __forceinline__ v4i ntload4(const int* p) {
  return __builtin_nontemporal_load((const v4i*)p);
}

// ---------------- K1: init deg (=1.0 for self loop) and agg1 (=0) --------
__global__ void gcn_init(float* __restrict__ deg, float* __restrict__ agg1, int n) {
  int i = blockIdx.x * blockDim.x + threadIdx.x;
  if (i < n) { deg[i] = 1.0f; agg1[i] = 0.0f; }
}

// ---------------- K2: degree scatter  deg[col] += 1 ----------------------
__global__ void gcn_degree(const int* __restrict__ col, float* __restrict__ deg, int e) {
  int i = blockIdx.x * blockDim.x + threadIdx.x;
  int base = i << 2;
  if (base + 3 < e) {
    v4i c = ntload4(col + base);
    atomicAdd(&deg[c.x], 1.0f);
    atomicAdd(&deg[c.y], 1.0f);
    atomicAdd(&deg[c.z], 1.0f);
    atomicAdd(&deg[c.w], 1.0f);
  } else {
    for (int j = base; j < e; ++j) atomicAdd(&deg[col[j]], 1.0f);
  }
}

// ---------------- K3: dis = rsqrt(deg) in place; xp = x*dis --------------
__global__ void gcn_dis(const float* __restrict__ x, float* __restrict__ deg_dis,
                        float* __restrict__ xp, int n) {
  int i = blockIdx.x * blockDim.x + threadIdx.x;
  if (i < n) {
    float d = rsqrtf(deg_dis[i]);   // deg >= 1 always (self loop)
    deg_dis[i] = d;
    xp[i] = x[i] * d;
  }
}

// ---------------- K4: layer-1 scalar scatter  agg1[col] += xp[row]*dis[col]
__global__ void gcn_l1(const int* __restrict__ row, const int* __restrict__ col,
                       const float* __restrict__ xp, const float* __restrict__ dis,
                       float* __restrict__ agg1, int e) {
  int i = blockIdx.x * blockDim.x + threadIdx.x;
  int base = i << 2;
  if (base + 3 < e) {
    v4i r = ntload4(row + base);
    v4i c = ntload4(col + base);
    atomicAdd(&agg1[c.x], xp[r.x] * dis[c.x]);
    atomicAdd(&agg1[c.y], xp[r.y] * dis[c.y]);
    atomicAdd(&agg1[c.z], xp[r.z] * dis[c.z]);
    atomicAdd(&agg1[c.w], xp[r.w] * dis[c.w]);
  } else {
    for (int j = base; j < e; ++j) {
      int rj = row[j], cj = col[j];
      atomicAdd(&agg1[cj], xp[rj] * dis[cj]);
    }
  }
}

// ---------------- K5: node transform via WMMA f32 16x16x4 ----------------
// Per wave: 16 nodes.  h1[m,j] = relu(a_m*W1[j]+b1[j]) (16 wide),
// h2 = h1 @ W2 (16x2) computed as 4 chained V_WMMA_F32_16X16X4_F32.
// Outputs: h2p[node][k] = h2*dis(node)  (pre-scaled by dis[row] for layer-2 edges)
//          out[node][k] = b2[k] + h2p*dis(node)  (self-loop message folded in)
__global__ void gcn_node(const float* __restrict__ agg1,
                         const float* __restrict__ xp,
                         const float* __restrict__ dis,
                         const float* __restrict__ W1,
                         const float* __restrict__ b1,
                         const float* __restrict__ W2,
                         const float* __restrict__ b2,
                         float* __restrict__ h2p,
                         float* __restrict__ out,
                         int n) {
  int lane = threadIdx.x & 31;
  int wave = threadIdx.x >> 5;
  int tile = blockIdx.x * (blockDim.x >> 5) + wave;
  int base = tile * 16;
  int m    = lane & 15;      // node within tile / B-column index
  int half = lane >> 4;      // K-pair selector per ISA f32 A/B layout
  int node = base + m;
  int nc   = node < n ? node : (n - 1);   // clamp loads; keep EXEC full for WMMA

  // total layer-1 aggregate incl. self loop: agg1 + x*dis^2 = agg1 + xp*dis
  float av = agg1[nc] + xp[nc] * dis[nc];

  // Column clamp/mask for B: only columns 0..1 of W2 are real; zero the rest
  // via VALU select+multiply (no exec-mask branching around the loads).
  int   wcol  = (m < 2) ? m : 0;
  float wmask = (m < 2) ? 1.0f : 0.0f;

  v8f acc = {};
#pragma unroll
  for (int s = 0; s < 4; ++s) {
    int j0 = s * 4 + half * 2;          // this lane's K indices: j0, j0+1 (even)
    // A (16x4 f32): lane m,half holds A[m][k0], A[m][k0+1]
    v2f w1 = *(const v2f*)(W1 + j0);
    v2f bi = *(const v2f*)(b1 + j0);
    v2f a;
    a.x = fmaxf(av * w1.x + bi.x, 0.0f);
    a.y = fmaxf(av * w1.y + bi.y, 0.0f);
    // B (4x16 f32): lane holds B[k0][m], B[k0+1][m]
    v2f bb;
    bb.x = W2[j0 * 2 + wcol]       * wmask;
    bb.y = W2[(j0 + 1) * 2 + wcol] * wmask;
    acc = __builtin_amdgcn_wmma_f32_16x16x4_f32(
        /*neg_a=*/false, a, /*neg_b=*/false, bb,
        /*c_mod=*/(short)0, acc, /*reuse_a=*/false, /*reuse_b=*/false);
  }

  // D (16x16 f32): lane n = lane&15, VGPR r holds M = r + half*8.
  if (m < 2) {
    float bv = b2[m];
#pragma unroll
    for (int r = 0; r < 8; ++r) {
      int nd = base + r + half * 8;
      if (nd < n) {
        float h2 = acc[r];
        float dv = dis[nd];
        float hp = h2 * dv;
        h2p[nd * 2 + m] = hp;            // pre-scaled message source
        out[nd * 2 + m] = bv + hp * dv;  // bias + self-loop term
      }
    }
  }
}

// ---------------- K6: layer-2 scatter  out[col] += h2p[row]*dis[col] -----
__global__ void gcn_l2(const int* __restrict__ row, const int* __restrict__ col,
                       const float* __restrict__ h2p, const float* __restrict__ dis,
                       float* __restrict__ out, int e) {
  int i = blockIdx.x * blockDim.x + threadIdx.x;
  int base = i << 2;
  if (base + 3 < e) {
    v4i r = ntload4(row + base);
    v4i c = ntload4(col + base);
#pragma unroll
    for (int j = 0; j < 4; ++j) {
      int rj = (j == 0) ? r.x : (j == 1) ? r.y : (j == 2) ? r.z : r.w;
      int cj = (j == 0) ? c.x : (j == 1) ? c.y : (j == 2) ? c.z : c.w;
      float dv = dis[cj];
      v2f hp = *(const v2f*)(h2p + 2 * (size_t)rj);
      atomicAdd(&out[2 * (size_t)cj + 0], hp.x * dv);
      atomicAdd(&out[2 * (size_t)cj + 1], hp.y * dv);
    }
  } else {
    for (int j = base; j < e; ++j) {
      int rj = row[j], cj = col[j];
      float dv = dis[cj];
      atomicAdd(&out[2 * (size_t)cj + 0], h2p[2 * (size_t)rj + 0] * dv);
      atomicAdd(&out[2 * (size_t)cj + 1], h2p[2 * (size_t)rj + 1] * dv);
    }
  }
}

extern "C" void kernel_launch(void* const* d_in, const int* in_sizes, int n_in,
                              void* d_out, int out_size, void* d_ws, size_t ws_size,
                              hipStream_t stream) {
  const float* x  = (const float*)d_in[0];
  const int*   ei = (const int*)d_in[1];
  const float* W1 = (const float*)d_in[2];
  const float* b1 = (const float*)d_in[3];
  const float* W2 = (const float*)d_in[4];
  const float* b2 = (const float*)d_in[5];
  float* out = (float*)d_out;

  const int n = in_sizes[0];       // x is (N,1)
  const int e = in_sizes[1] / 2;   // edge_index is (2,E)
  const int* row = ei;             // edge_index[0]
  const int* col = ei + e;         // edge_index[1]

  // workspace: dis | xp | agg1 | h2p(2N)   (np padded for 16B alignment)
  size_t np   = (size_t)((n + 3) & ~3);
  float* ws   = (float*)d_ws;
  float* dis  = ws;
  float* xp   = ws + np;
  float* agg1 = ws + 2 * np;
  float* h2p  = ws + 3 * np;

  const int tb = 256;
  const int nb_node = (n + tb - 1) / tb;
  const int ethreads = (e + 3) / 4;
  const int nb_edge = (ethreads + tb - 1) / tb;
  const int tiles = (n + 15) / 16;
  const int nb_tile = (tiles + 7) / 8;   // 8 waves (tiles) per 256-thread block

  gcn_init  <<<nb_node, tb, 0, stream>>>(dis, agg1, n);
  gcn_degree<<<nb_edge, tb, 0, stream>>>(col, dis, e);
  gcn_dis   <<<nb_node, tb, 0, stream>>>(x, dis, xp, n);
  gcn_l1    <<<nb_edge, tb, 0, stream>>>(row, col, xp, dis, agg1, e);
  gcn_node  <<<nb_tile, 256, 0, stream>>>(agg1, xp, dis, W1, b1, W2, b2, h2p, out, n);
  gcn_l2    <<<nb_edge, tb, 0, stream>>>(row, col, h2p, dis, out, e);
}